// IOTransformer_4440996184120
// MI455X (gfx1250) — compile-verified
//
#include <hip/hip_runtime.h>
#include <math.h>

typedef float v2f __attribute__((ext_vector_type(2)));
typedef float v8f __attribute__((ext_vector_type(8)));

#define TT 2048
#define DD 256
#define NBATCH 8
#define NACT 64
#define NTIME 32
#define NTOT 96
#define ROWS (NBATCH * TT)            // 16384
#define OUT_TIME_OFF (ROWS * NACT)    // 1048576
#define PSTRIDE 112                   // float2 per k-pair row: +224 words => +32 banks
#define KCHUNK 64

__device__ __forceinline__ float softplus_f(float x) { return log1pf(expf(x)); }

// ---------------------------------------------------------------------------
// Kernel 1: fused GEMM  out = h @ [Wn + s_ta*E_act ; Wt + s_tt*E_time]^T + bias
// wave32; each wave computes a 16(M) x 96(N) strip via 6 f32 WMMA accumulators.
// LDS holds the combined weight chunk as k-PAIRS (float2) so each B fragment
// is a single conflict-free ds_load_b64 landing in an even-aligned VGPR pair.
// ---------------------------------------------------------------------------
__global__ __launch_bounds__(256)
void gemm_heads_kernel(const float* __restrict__ h,
                       const float* __restrict__ E,
                       const float* __restrict__ Wn,
                       const float* __restrict__ bn,
                       const float* __restrict__ Wt,
                       const float* __restrict__ bt,
                       const float* __restrict__ tsa,
                       const float* __restrict__ tst,
                       float* __restrict__ out) {
  __shared__ v2f lw2[(KCHUNK / 2) * PSTRIDE];   // 32 pair-rows * 112 * 8B = 28 KB

  const int tid   = threadIdx.x;
  const int wave  = tid >> 5;
  const int lane  = tid & 31;
  const int lhalf = lane >> 4;                  // 0 or 1 (half-wave)
  const int l16   = lane & 15;
  const int m_base = blockIdx.x * 128 + wave * 16;

  const float s_ta = softplus_f(tsa[0]);
  const float s_tt = softplus_f(tst[0]);

  // Accumulators initialized with the bias (bias depends only on N = column).
  v8f acc[6];
#pragma unroll
  for (int j = 0; j < 6; ++j) {
    int n = j * 16 + l16;
    float bv = (n < NACT) ? bn[n] : bt[n - NACT];
#pragma unroll
    for (int r = 0; r < 8; ++r) acc[j][r] = bv;
  }

  float* lwf = (float*)lw2;

  for (int kc = 0; kc < DD; kc += KCHUNK) {
    __syncthreads();
    // Stage combined-weight chunk: lw2[k>>1][n] = {W_comb[n][kc+k], W_comb[n][kc+k+1]}
    // Mapping keeps global reads coalesced along k.
#pragma unroll
    for (int i = 0; i < (KCHUNK * NTOT) / 256; ++i) {   // 24 iters
      int idx = i * 256 + tid;
      int kk  = idx & (KCHUNK - 1);
      int n   = idx >> 6;
      int kg  = kc + kk;
      float wv;
      if (n < NACT) wv = Wn[n * DD + kg] + s_ta * E[(4 + n) * DD + kg];
      else          wv = Wt[(n - NACT) * DD + kg] + s_tt * E[(68 + (n - NACT)) * DD + kg];
      lwf[((kk >> 1) * PSTRIDE + n) * 2 + (kk & 1)] = wv;
    }
    __syncthreads();

#pragma unroll 4
    for (int ks = 0; ks < KCHUNK; ks += 4) {
      // A fragment (16x4 f32): lanes 0-15 K=k0,k0+1 ; lanes 16-31 K=k0+2,k0+3
      int kk = kc + ks + (lhalf << 1);
      v2f a = *(const v2f*)(h + (m_base + l16) * DD + kk);
      int p = (ks >> 1) + lhalf;                // k-pair row for this half-wave
#pragma unroll
      for (int j = 0; j < 6; ++j) {
        int n = j * 16 + l16;
        v2f b = lw2[p * PSTRIDE + n];           // single ds_load_b64, even pair
        acc[j] = __builtin_amdgcn_wmma_f32_16x16x4_f32(
            false, a, false, b, (short)0, acc[j], false, false);
      }
    }
  }

  // Store: C/D layout — VGPR r: lanes 0-15 (M=r, N=lane), lanes 16-31 (M=8+r)
#pragma unroll
  for (int j = 0; j < 6; ++j) {
    int n = j * 16 + l16;
#pragma unroll
    for (int r = 0; r < 8; ++r) {
      int row = m_base + r + (lhalf << 3);
      if (n < NACT) out[row * NACT + n] = acc[j][r];
      else          out[OUT_TIME_OFF + row * NTIME + (n - NACT)] = acc[j][r];
    }
  }
}

// ---------------------------------------------------------------------------
// Kernel 2: causal prototype head. One block per (batch row, 32-class segment).
// seg 0: act classes 0..31, seg 1: act classes 32..63, seg 2: time classes.
// A_c = Hsum_c + alpha*Enorm_c kept in LDS; direction-preserving so per-class
// counts are unnecessary (division by positive scalar cancels in l2norm).
// Adds sp(scale)*sp(temp)*cos(h_t, A_c) into d_out at valid label positions.
// ---------------------------------------------------------------------------
__global__ __launch_bounds__(256)
void proto_kernel(const int* __restrict__ tokens,
                  const float* __restrict__ h,
                  const float* __restrict__ E,
                  const float* __restrict__ psa, const float* __restrict__ pst,
                  const float* __restrict__ ppa, const float* __restrict__ ppt,
                  const float* __restrict__ pta, const float* __restrict__ ptt,
                  float* __restrict__ out) {
  __shared__ float A[32 * 256];     // 32 KB prototype accumulators
  __shared__ float hbuf[256];
  __shared__ float red[256];
  __shared__ float invn[32];
  __shared__ int   evbuf[2048];
  __shared__ int   nev_sh;

  const int b   = blockIdx.x / 3;
  const int seg = blockIdx.x % 3;
  const bool is_time = (seg == 2);
  const int lo = is_time ? 68 : 4;
  const int hi = is_time ? 100 : 68;
  const int c0 = (seg == 1) ? 32 : 0;
  const int eb = lo + c0;                       // E row base for this segment

  const float alpha = softplus_f(is_time ? ppt[0] : ppa[0]);
  const float st    = softplus_f(is_time ? pst[0] : psa[0]) *
                      softplus_f(is_time ? ptt[0] : pta[0]);

  const int tid  = threadIdx.x;
  const int wave = tid >> 5;
  const int lane = tid & 31;

  // Event list: one entry per label position that supports or emits.
  if (tid == 0) {
    int nsup = 0, nev = 0;
    const int* tk = tokens + b * TT;
    for (int t = 0; t < TT; ++t) {
      if (tk[t] == 3) {
        int nxt  = tk[(t + 1) & (TT - 1)];
        int sup  = (nxt >= lo && nxt < hi) ? 1 : 0;
        int emit = (nsup > 0) ? 1 : 0;
        if (sup | emit)
          evbuf[nev++] = (t << 9) | (emit << 8) | (sup ? (nxt - lo + 1) : 0);
        nsup += sup;
      }
    }
    nev_sh = nev;
  }

  // Per-class inverse norms of detached E rows.
  if (tid < 32) {
    float s = 0.f;
    const float* er = E + (eb + tid) * DD;
    for (int d = 0; d < DD; ++d) { float v = er[d]; s += v * v; }
    invn[tid] = 1.0f / fmaxf(sqrtf(s), 1e-12f);
  }
  __syncthreads();

  // A init = alpha * Enorm
  for (int i = 0; i < 32; ++i) {
    int idx = i * 256 + tid;
    int c = idx >> 8, d = idx & 255;
    A[idx] = alpha * E[(eb + c) * DD + d] * invn[c];
  }
  __syncthreads();

  const int nev = nev_sh;
  for (int e = 0; e < nev; ++e) {
    int ev   = evbuf[e];
    int t    = ev >> 9;
    int emit = (ev >> 8) & 1;
    int clsp = ev & 255;                    // class+1 within head, 0 = no support
    int r    = b * TT + t;

    float hv = h[r * DD + tid];
    hbuf[tid] = hv;
    red[tid]  = hv * hv;
    __syncthreads();
    for (int s = 128; s > 0; s >>= 1) {
      if (tid < s) red[tid] += red[tid + s];
      __syncthreads();
    }
    float invh = 1.0f / fmaxf(sqrtf(red[0]), 1e-12f);

    if (emit) {
      // each wave handles 4 classes; lane covers 8 contiguous d-elements
#pragma unroll
      for (int q = 0; q < 4; ++q) {
        int c = wave * 4 + q;
        const float* ac = A + c * 256;
        float dot = 0.f, asq = 0.f;
#pragma unroll
        for (int u = 0; u < 8; ++u) {
          float av = ac[lane * 8 + u];
          float hx = hbuf[lane * 8 + u];
          dot += av * hx;
          asq += av * av;
        }
#pragma unroll
        for (int off = 16; off > 0; off >>= 1) {
          dot += __shfl_xor(dot, off, 32);
          asq += __shfl_xor(asq, off, 32);
        }
        if (lane == 0) {
          float sim = dot * invh / fmaxf(sqrtf(asq), 1e-12f);
          float add = st * sim;
          if (!is_time) out[r * NACT + (c0 + c)] += add;
          else          out[OUT_TIME_OFF + r * NTIME + c] += add;
        }
      }
    }
    __syncthreads();

    if (clsp) {                              // support update: Hsum_c += h_norm
      int cl = clsp - 1 - c0;
      if (cl >= 0 && cl < 32) A[cl * 256 + tid] += hbuf[tid] * invh;
    }
    __syncthreads();
  }
}

// ---------------------------------------------------------------------------
extern "C" void kernel_launch(void* const* d_in, const int* in_sizes, int n_in,
                              void* d_out, int out_size, void* d_ws, size_t ws_size,
                              hipStream_t stream) {
  const int*   tokens = (const int*)  d_in[0];
  const float* h      = (const float*)d_in[1];
  const float* E      = (const float*)d_in[2];
  const float* Wn     = (const float*)d_in[3];
  const float* bn     = (const float*)d_in[4];
  const float* Wt     = (const float*)d_in[5];
  const float* bt     = (const float*)d_in[6];
  const float* tsa    = (const float*)d_in[7];
  const float* tst    = (const float*)d_in[8];
  const float* psa    = (const float*)d_in[9];
  const float* pst    = (const float*)d_in[10];
  const float* ppa    = (const float*)d_in[11];
  const float* ppt    = (const float*)d_in[12];
  const float* pta    = (const float*)d_in[13];
  const float* ptt    = (const float*)d_in[14];
  float* out = (float*)d_out;

  gemm_heads_kernel<<<ROWS / 128, 256, 0, stream>>>(h, E, Wn, bn, Wt, bt, tsa, tst, out);
  proto_kernel<<<NBATCH * 3, 256, 0, stream>>>(tokens, h, E, psa, pst, ppa, ppt,
                                               pta, ptt, out);
}